// GATLayer_31559419691754
// MI455X (gfx1250) — compile-verified
//
#include <hip/hip_runtime.h>
#include <math.h>

#define IN_CH    128
#define HEADS    4
#define OUT_CH   16
#define HC       64          // HEADS * OUT_CH
#define NEG_SLOPE 0.2f

typedef __attribute__((ext_vector_type(2))) float        v2f;
typedef __attribute__((ext_vector_type(8))) float        v8f;
typedef __attribute__((ext_vector_type(4))) unsigned int v4u;
typedef __attribute__((ext_vector_type(8))) int          v8i;
typedef __attribute__((ext_vector_type(4))) int          v4i;

// ---------------------------------------------------------------------------
// z = x @ W via V_WMMA_F32_16X16X4_F32.
// One wave -> one 16-row x 64-col strip (4 accumulators, 128 WMMAs).
// W (128x64 f32, 32 KB) is DMA'd into LDS once per block by the Tensor Data
// Mover (tensor_load_to_lds + s_wait_tensorcnt), then B fragments come from
// LDS. A fragments are read once per wave and feed 4 independent WMMA chains.
// ---------------------------------------------------------------------------
__global__ __launch_bounds__(128) void gat_gemm_wmma(
    const float* __restrict__ x, const float* __restrict__ Wm,
    float* __restrict__ z, int nv) {
  __shared__ float Wlds[IN_CH * HC];       // 32 KB

  const int wave = threadIdx.x >> 5;
  const int lane = threadIdx.x & 31;

  // ---- Stage W into LDS -------------------------------------------------
#if __has_builtin(__builtin_amdgcn_tensor_load_to_lds)
  if (wave == 0) {
    const unsigned lds_off = (unsigned)(uintptr_t)(&Wlds[0]); // ISA: LDS addr = flat[31:0]
    const unsigned long long ga = (unsigned long long)(uintptr_t)Wm;
    v4u g0;
    g0.x = 1u;                                            // count=1, user descriptor
    g0.y = lds_off;                                       // lds_addr (bytes)
    g0.z = (unsigned)(ga & 0xffffffffu);                  // global_addr[31:0]
    g0.w = (unsigned)((ga >> 32) & 0x01ffffffu)           // global_addr[56:32]
           | 0x80000000u;                                 // type=2 ("image")
    v8i g1;
    g1[0] = (2 << 16);                                    // data_size=4B; no mask/pad
    g1[1] = (HC & 0xffff) << 16;                          // tensor_dim0 = 64   [63:48]
    g1[2] = ((IN_CH & 0xffff) << 16);                     // td0 hi=0 | tensor_dim1=128 lo
    g1[3] = (HC << 16);                                   // td1 hi=0 | tile_dim0 = 64
    g1[4] = IN_CH;                                        // tile_dim1 = 128; tile_dim2 = 0
    g1[5] = HC;                                           // tensor_dim0_stride = 64 (lo32)
    g1[6] = 0;                                            // stride0 hi | stride1 lo
    g1[7] = 0;
    v4i gz4 = {0, 0, 0, 0};
    v8i gz8 = {0, 0, 0, 0, 0, 0, 0, 0};
    __builtin_amdgcn_tensor_load_to_lds(g0, g1, gz4, gz4, gz8, 0);
#if __has_builtin(__builtin_amdgcn_s_wait_tensorcnt)
    __builtin_amdgcn_s_wait_tensorcnt(0);
#else
    asm volatile("s_wait_tensorcnt 0x0" ::: "memory");
#endif
  }
#else
  // Fallback: cooperative vector copy.
  for (int i = threadIdx.x; i < IN_CH * HC / 4; i += 128) {
    ((float4*)Wlds)[i] = ((const float4*)Wm)[i];
  }
#endif
  __syncthreads();   // all threads reach this before any early exit

  // ---- Compute ----------------------------------------------------------
  const int rowTiles = (nv + 15) >> 4;
  const int rowTile = blockIdx.x * 4 + wave;
  if (rowTile >= rowTiles) return;          // wave-uniform branch

  const int row0 = rowTile << 4;
  const int lm = lane & 15;                  // M (A) / N (B) index
  const int hi = lane >> 4;                  // K-half selector

  int arow = row0 + lm;
  if (arow >= nv) arow = nv - 1;             // clamp (select, no divergence)
  const float* xrow = x + (size_t)arow * IN_CH + 2 * hi;

  v8f acc0 = {}, acc1 = {}, acc2 = {}, acc3 = {};
  #pragma unroll
  for (int k = 0; k < IN_CH; k += 4) {
    // A 16x4: lanes 0-15 hold K = k,k+1 ; lanes 16-31 hold K = k+2,k+3
    v2f a = *(const v2f*)(xrow + k);
    const int kk = k + 2 * hi;
    const float* w0 = &Wlds[(size_t)kk * HC + lm];
    const float* w1 = w0 + HC;
    v2f b0 = {w0[0],  w1[0]};
    v2f b1 = {w0[16], w1[16]};
    v2f b2 = {w0[32], w1[32]};
    v2f b3 = {w0[48], w1[48]};
    acc0 = __builtin_amdgcn_wmma_f32_16x16x4_f32(false, a, false, b0, (short)0, acc0, false, false);
    acc1 = __builtin_amdgcn_wmma_f32_16x16x4_f32(false, a, false, b1, (short)0, acc1, false, false);
    acc2 = __builtin_amdgcn_wmma_f32_16x16x4_f32(false, a, false, b2, (short)0, acc2, false, false);
    acc3 = __builtin_amdgcn_wmma_f32_16x16x4_f32(false, a, false, b3, (short)0, acc3, false, false);
  }

  // C/D layout: VGPR r -> M = row0 + 8*hi + r, N = 16*tile + lm
  const int mbase = row0 + hi * 8;
  #pragma unroll
  for (int r = 0; r < 8; ++r) {
    const int m = mbase + r;
    if (m < nv) {
      float* zr = z + (size_t)m * HC + lm;
      zr[0]  = acc0[r];
      zr[16] = acc1[r];
      zr[32] = acc2[r];
      zr[48] = acc3[r];
    }
  }
}

// ---------------------------------------------------------------------------
// Per-node attention logits + init of segment-max / denom buffers.
// ---------------------------------------------------------------------------
__global__ void gat_node_logits(const float* __restrict__ z,
                                const float* __restrict__ a_src,
                                const float* __restrict__ a_dst,
                                float* __restrict__ es, float* __restrict__ ed,
                                float* __restrict__ m, float* __restrict__ denom,
                                int nv) {
  const int t = blockIdx.x * blockDim.x + threadIdx.x;
  if (t >= nv * HEADS) return;
  const int v = t >> 2, h = t & 3;
  const float* zr = z + (size_t)v * HC + h * OUT_CH;
  const float* as = a_src + h * OUT_CH;
  const float* ad = a_dst + h * OUT_CH;
  float s = 0.f, d = 0.f;
  #pragma unroll
  for (int c = 0; c < OUT_CH; ++c) {
    const float zv = zr[c];
    s = fmaf(zv, as[c], s);
    d = fmaf(zv, ad[c], d);
  }
  es[t] = s;
  ed[t] = d;
  m[t] = -3.0e38f;   // acts as -inf; only nodes with incoming edges are read
  denom[t] = 0.f;
}

__device__ __forceinline__ void atomicMaxF(float* addr, float val) {
  int* ai = (int*)addr;
  int cur = __float_as_int(*addr);
  while (__int_as_float(cur) < val) {
    const int assumed = cur;
    cur = atomicCAS(ai, assumed, __float_as_int(val));
    if (cur == assumed) break;
  }
}

// Pass 1: e = leaky_relu(es[src] + ed[dst]); segment max into m[dst].
__global__ void gat_edge_max(const int* __restrict__ src, const int* __restrict__ dst,
                             const float* __restrict__ es, const float* __restrict__ ed,
                             float* __restrict__ e_ws, float* __restrict__ m, int ne) {
  const int t = blockIdx.x * blockDim.x + threadIdx.x;
  if (t >= ne * HEADS) return;
  const int e = t >> 2, h = t & 3;
  const int s = src[e], d = dst[e];
  float v = es[s * HEADS + h] + ed[d * HEADS + h];
  v = v > 0.f ? v : NEG_SLOPE * v;
  e_ws[t] = v;
  atomicMaxF(&m[d * HEADS + h], v);
}

// Pass 2: exp_e = exp(e - m[dst]); segment sum into denom[dst].
__global__ void gat_edge_exp(const int* __restrict__ dst, const float* __restrict__ m,
                             float* __restrict__ e_ws, float* __restrict__ denom, int ne) {
  const int t = blockIdx.x * blockDim.x + threadIdx.x;
  if (t >= ne * HEADS) return;
  const int e = t >> 2, h = t & 3;
  const int d = dst[e];
  const float ex = expf(e_ws[t] - m[d * HEADS + h]);
  e_ws[t] = ex;
  atomicAdd(&denom[d * HEADS + h], ex);
}

// Pass 3: alpha = exp_e / (denom + 1e-9); agg[dst] += z[src] * alpha.
__global__ void gat_edge_agg(const int* __restrict__ src, const int* __restrict__ dst,
                             const float* __restrict__ z, const float* __restrict__ e_ws,
                             const float* __restrict__ denom, float* __restrict__ agg,
                             int ne) {
  const int t = blockIdx.x * blockDim.x + threadIdx.x;
  if (t >= ne * HEADS) return;
  const int e = t >> 2, h = t & 3;
  const int s = src[e], d = dst[e];
  const float alpha = e_ws[t] / (denom[d * HEADS + h] + 1e-9f);
  const float4* zr = (const float4*)(z + (size_t)s * HC + h * OUT_CH);
  float* ag = agg + (size_t)d * HC + h * OUT_CH;
  #pragma unroll
  for (int q = 0; q < 4; ++q) {
    const float4 zv = zr[q];
    atomicAdd(&ag[4 * q + 0], zv.x * alpha);
    atomicAdd(&ag[4 * q + 1], zv.y * alpha);
    atomicAdd(&ag[4 * q + 2], zv.z * alpha);
    atomicAdd(&ag[4 * q + 3], zv.w * alpha);
  }
}

// Final: out = elu(agg)
__global__ void gat_elu(const float* __restrict__ agg, float* __restrict__ out, int n) {
  const int t = blockIdx.x * blockDim.x + threadIdx.x;
  if (t >= n) return;
  const float v = agg[t];
  out[t] = v > 0.f ? v : (expf(v) - 1.f);
}

extern "C" void kernel_launch(void* const* d_in, const int* in_sizes, int n_in,
                              void* d_out, int out_size, void* d_ws, size_t ws_size,
                              hipStream_t stream) {
  const float* x        = (const float*)d_in[0];
  const int*   edge_idx = (const int*)d_in[1];   // [2,E]: row0=src, row1=dst
  const float* W        = (const float*)d_in[2];
  const float* a_src    = (const float*)d_in[3];
  const float* a_dst    = (const float*)d_in[4];

  const int nv = in_sizes[0] / IN_CH;
  const int ne = in_sizes[1] / 2;
  const int* src = edge_idx;
  const int* dst = edge_idx + ne;

  // Workspace layout (floats)
  float* ws    = (float*)d_ws;
  float* z     = ws;                          // nv*64
  float* es    = z     + (size_t)nv * HC;     // nv*4
  float* ed    = es    + (size_t)nv * HEADS;  // nv*4
  float* m     = ed    + (size_t)nv * HEADS;  // nv*4
  float* denom = m     + (size_t)nv * HEADS;  // nv*4
  float* e_ws  = denom + (size_t)nv * HEADS;  // ne*4
  float* agg   = e_ws  + (size_t)ne * HEADS;  // nv*64
  (void)ws_size; (void)n_in; (void)out_size;

  // Zero the aggregation buffer (graph-capture-safe async memset)
  (void)hipMemsetAsync(agg, 0, (size_t)nv * HC * sizeof(float), stream);

  // 1) z = x @ W  (TDM-staged W + WMMA fp32; 1 wave = 16 rows x 64 cols)
  const int rowTiles = (nv + 15) >> 4;
  gat_gemm_wmma<<<(rowTiles + 3) / 4, 128, 0, stream>>>(x, W, z, nv);

  // 2) per-node logits + init
  const int nvh = nv * HEADS;
  gat_node_logits<<<(nvh + 255) / 256, 256, 0, stream>>>(z, a_src, a_dst, es, ed, m, denom, nv);

  // 3..5) edge passes
  const int neh = ne * HEADS;
  const int egrid = (neh + 255) / 256;
  gat_edge_max<<<egrid, 256, 0, stream>>>(src, dst, es, ed, e_ws, m, ne);
  gat_edge_exp<<<egrid, 256, 0, stream>>>(dst, m, e_ws, denom, ne);
  gat_edge_agg<<<egrid, 256, 0, stream>>>(src, dst, z, e_ws, denom, agg, ne);

  // 6) ELU -> output
  const int nout = nv * HC;
  gat_elu<<<(nout + 255) / 256, 256, 0, stream>>>(agg, (float*)d_out, nout);
}